// CDC_CA_CB_conv_72404558676429
// MI455X (gfx1250) — compile-verified
//
#include <hip/hip_runtime.h>

// ---- CDNA5 WMMA types ----
typedef __attribute__((ext_vector_type(2))) float v2f;  // A/B frag: 16x4 / 4x16 f32 = 2 VGPRs
typedef __attribute__((ext_vector_type(8))) float v8f;  // C/D frag: 16x16 f32 = 8 VGPRs

#define THETA   0.7f
#define BATCH   32
#define CH      128               // Cin == Cout
#define HW      128               // H == W
#define KDIM    (CH*9)            // 1152
#define TILE_W  64                // pixels per block tile (4 WMMA N-tiles per wave)
#define XS_W    (TILE_W + 2)      // 66 (halo)
#define XS_ROW  (3*XS_W)          // 198 floats per channel (3 rows x 66 cols)
#define XS_ELEMS (CH*XS_ROW)      // 25344 floats = 99 KB LDS

__global__ __launch_bounds__(256)
void cdc_ca_conv_wmma_f32(const float* __restrict__ x,
                          const float* __restrict__ Wt,
                          const float* __restrict__ bias,
                          float* __restrict__ out) {
    __shared__ float xs[XS_ELEMS];     // zero-padded x tile: [ci][dy(3)][dx(66)]
    __shared__ float stile[XS_ROW];    // channel sums at the 3x66 positions
    __shared__ float diffv[TILE_W];    // laplacian per output pixel
    __shared__ float sbias[CH];

    const int tid = threadIdx.x;
    const int blk = blockIdx.x;
    const int b   = blk >> 8;             // / (128 rows * 2 wtiles)
    const int rem = blk & 255;
    const int h   = rem >> 1;
    const int w0  = (rem & 1) * TILE_W;

    // ---- stage x tile (zero-padded halo == conv zero padding) + bias ----
    const int xbase = b * CH * HW * HW;
    for (int e = tid; e < XS_ELEMS; e += 256) {
        const int ci = e / XS_ROW;
        const int r2 = e - ci * XS_ROW;
        const int dy = r2 / XS_W;
        const int dx = r2 - dy * XS_W;
        const int gh = h + dy - 1;
        const int gw = w0 + dx - 1;
        float v = 0.0f;
        if ((unsigned)gh < (unsigned)HW && (unsigned)gw < (unsigned)HW)
            v = x[xbase + (ci * HW + gh) * HW + gw];
        xs[e] = v;
    }
    if (tid < CH) sbias[tid] = bias[tid];
    __syncthreads();

    // ---- channel sums for the 5-point stencil (valid wherever in-image) ----
    if (tid < XS_ROW) {
        float s = 0.0f;
        #pragma unroll 8
        for (int ci = 0; ci < CH; ++ci) s += xs[ci * XS_ROW + tid];
        stile[tid] = s;
    }
    __syncthreads();

    // ---- laplacian with edge replication: clamped neighbors stay inside tile ----
    if (tid < TILE_W) {
        const int p = tid;
        const int w = w0 + p;
        const float c  = stile[XS_W + p + 1];
        const float up = (h > 0)      ? stile[p + 1]            : c;
        const float dn = (h < HW - 1) ? stile[2 * XS_W + p + 1] : c;
        const float lf = (w > 0)      ? stile[XS_W + p]         : c;
        const float rt = (w < HW - 1) ? stile[XS_W + p + 2]     : c;
        diffv[p] = up + dn + lf + rt - 4.0f * c;
    }
    __syncthreads();

    // ---- implicit GEMM: per wave M=16 co, N=64 pixels (4 tiles), K=1152 by 4 ----
    // A fragment (weights) is pixel-independent -> reused across all 4 N tiles:
    // 0.5 global VMEM + 1 ds_load_2addr per v_wmma_f32_16x16x4_f32.
    const int lane   = tid & 31;
    const int wave   = tid >> 5;
    const int hi     = lane >> 4;   // half-wave: K pair select / co offset +8
    const int m      = lane & 15;   // A row (co) and B col (pixel within tile)
    const int coBase = wave * 16;

    v8f acc0 = {}, acc1 = {}, acc2 = {}, acc3 = {};

    // K order: kk = r*128 + ci  (r = kh*3+kw) -> compile-time-constant strides.
    for (int r = 0; r < 9; ++r) {
        const int kh = r / 3;
        const int kw = r - kh * 3;
        // A: lane holds W[coBase+m][ci = 2*hi + 4*s + {0,1}][r]; ci stride = 9 floats
        const float* aW  = Wt + (coBase + m) * KDIM + r + 18 * hi;
        // B: lane holds xs[ci = 2*hi + 4*s + {0,1}][kh][kw + m (+16/+32/+48)]
        const float* bX  = &xs[(2 * hi) * XS_ROW + kh * XS_W + kw + m];
        __builtin_prefetch(aW + 288, 0, 0);   // 32 ci ahead in this r's weight stream
        #pragma unroll 4
        for (int s = 0; s < CH / 4; ++s) {
            v2f a, b0, b1, b2, b3;
            a.x  = aW[0];
            a.y  = aW[9];                // ci+1
            b0.x = bX[0];       b0.y = bX[XS_ROW];
            b1.x = bX[16];      b1.y = bX[XS_ROW + 16];
            b2.x = bX[32];      b2.y = bX[XS_ROW + 32];
            b3.x = bX[48];      b3.y = bX[XS_ROW + 48];
            acc0 = __builtin_amdgcn_wmma_f32_16x16x4_f32(
                false, a, false, b0, (short)0, acc0, false, false);
            acc1 = __builtin_amdgcn_wmma_f32_16x16x4_f32(
                false, a, false, b1, (short)0, acc1, false, false);
            acc2 = __builtin_amdgcn_wmma_f32_16x16x4_f32(
                false, a, false, b2, (short)0, acc2, false, false);
            acc3 = __builtin_amdgcn_wmma_f32_16x16x4_f32(
                false, a, false, b3, (short)0, acc3, false, false);
            aW += 36;            // ci += 4
            bX += 4 * XS_ROW;    // ci += 4
        }
    }

    // ---- epilogue: + bias[co] - THETA * diff[pixel] ----
    const float dadj0 = -THETA * diffv[m];
    const float dadj1 = -THETA * diffv[m + 16];
    const float dadj2 = -THETA * diffv[m + 32];
    const float dadj3 = -THETA * diffv[m + 48];
    const int   ob    = (b * CH * HW + h) * HW + w0 + m;
    #pragma unroll
    for (int v = 0; v < 8; ++v) {
        const int co  = coBase + v + 8 * hi;   // C/D layout: VGPR v, half-wave hi
        const int oix = ob + co * HW * HW;
        out[oix]      = acc0[v] + sbias[co] + dadj0;
        out[oix + 16] = acc1[v] + sbias[co] + dadj1;
        out[oix + 32] = acc2[v] + sbias[co] + dadj2;
        out[oix + 48] = acc3[v] + sbias[co] + dadj3;
    }
}

extern "C" void kernel_launch(void* const* d_in, const int* in_sizes, int n_in,
                              void* d_out, int out_size, void* d_ws, size_t ws_size,
                              hipStream_t stream) {
    (void)in_sizes; (void)n_in; (void)d_ws; (void)ws_size; (void)out_size;
    const float* x    = (const float*)d_in[0];
    const float* Wt   = (const float*)d_in[1];
    const float* bias = (const float*)d_in[2];
    float*       out  = (float*)d_out;

    const int blocks = BATCH * HW * (HW / TILE_W);  // 32 * 128 * 2 = 8192
    cdc_ca_conv_wmma_f32<<<dim3(blocks), dim3(256), 0, stream>>>(x, Wt, bias, out);
}